// ABlock_38594576122518
// MI455X (gfx1250) — compile-verified
//
#include <hip/hip_runtime.h>
#include <hip/hip_bf16.h>

// ---------------------------------------------------------------------------
// MI455X (gfx1250, wave32) implementation.
// Only F0 (1x1 conv), its attention map a0, and the final folded 3x3 conv
// matter (argsort over a size-1 axis in the reference selects index 0 always).
// K1/K3 use V_WMMA_F32_16X16X4_F32; K2/K3 stage tiles into LDS with
// GLOBAL_LOAD_ASYNC_TO_LDS_B32 (ASYNCcnt) when the toolchain exposes it.
// ---------------------------------------------------------------------------

typedef __attribute__((ext_vector_type(2))) float v2f;
typedef __attribute__((ext_vector_type(8))) float v8f;

#define B_    32
#define HW    65536          // 256*256
#define W_    256

#if defined(__gfx1250__) && __has_builtin(__builtin_amdgcn_global_load_async_to_lds_b32)
#define HAVE_ASYNC_LDS 1
#else
#define HAVE_ASYNC_LDS 0
#endif

// async gather: one 4B element global -> LDS, tracked by ASYNCcnt (no VGPR data)
__device__ __forceinline__ void ldgsts32(const float* g, float* l) {
#if HAVE_ASYNC_LDS
  __builtin_amdgcn_global_load_async_to_lds_b32(
      (__attribute__((address_space(1))) int*)(g),
      (__attribute__((address_space(3))) int*)(l), 0, 0);
#else
  *l = *g;
#endif
}

__device__ __forceinline__ void async_wait_all() {
#if HAVE_ASYNC_LDS
  asm volatile("s_wait_asynccnt 0" ::: "memory");
#endif
}

__device__ __forceinline__ v8f wmma4(v2f a, v2f b, v8f c) {
  // D = A(16x4 f32) * B(4x16 f32) + C(16x16 f32)
  return __builtin_amdgcn_wmma_f32_16x16x4_f32(
      /*neg_a=*/false, a, /*neg_b=*/false, b,
      /*c_mod=*/(short)0, c, /*reuse_a=*/false, /*reuse_b=*/false);
}

// ===========================================================================
// Kernel 1: F0 = w1(16x32) @ x + b1  (per-pixel), plus channel mean of F0.
// One wave handles one image row (256 px) = 16 WMMA tiles of 16 px.
// ===========================================================================
__global__ __launch_bounds__(256) void k1_f0_mean(
    const float* __restrict__ x, const float* __restrict__ w1,
    const float* __restrict__ b1, float* __restrict__ F0,
    float* __restrict__ meanm) {
  const int lane = threadIdx.x & 31;
  const int wave = threadIdx.x >> 5;
  const int gr   = blockIdx.x * 8 + wave;   // global row over all images
  const int b    = gr >> 8;
  const int row  = gr & 255;
  const int half = lane >> 4;               // lanes 0-15: K0/K1, 16-31: K2/K3
  const int l16  = lane & 15;

  // A (weights): lane m = l16; K pair = 4*kb + 2*half
  v2f A[8];
  const float* wrow = w1 + l16 * 32 + half * 2;
#pragma unroll
  for (int kb = 0; kb < 8; ++kb) {
    A[kb].x = wrow[4 * kb + 0];
    A[kb].y = wrow[4 * kb + 1];
  }
  // C/D layout: reg r holds channel r (lanes 0-15) or r+8 (lanes 16-31)
  v8f cb;
#pragma unroll
  for (int r = 0; r < 8; ++r) cb[r] = b1[r + half * 8];

  const float* xb = x + ((size_t)b * 32) * HW + row * W_;
  float* fb = F0 + ((size_t)b * 16) * HW + row * W_;
  float* mb = meanm + (size_t)b * HW + row * W_;

  for (int t = 0; t < 16; ++t) {
    const int col = t * 16 + l16;           // pixel N = l16 within tile
    v8f acc = cb;
#pragma unroll
    for (int kb = 0; kb < 8; ++kb) {
      const int c0 = 4 * kb + half * 2;     // input channel for B rows
      v2f bv;
      bv.x = xb[(size_t)c0 * HW + col];
      bv.y = xb[(size_t)(c0 + 1) * HW + col];
      acc = wmma4(A[kb], bv, acc);
    }
    // store F0 planar (coalesced: 16 lanes x 4B contiguous per channel)
#pragma unroll
    for (int r = 0; r < 8; ++r)
      fb[(size_t)(r + half * 8) * HW + col] = acc[r];
    // channel mean: regs sum = 8 channels; pair lanes l <-> l+16
    float s = 0.f;
#pragma unroll
    for (int r = 0; r < 8; ++r) s += acc[r];
    s += __shfl_xor(s, 16, 32);
    if (half == 0) mb[col] = s * (1.0f / 16.0f);
  }
}

// ===========================================================================
// Kernel 2: a0 = sigmoid(conv3x3(relu(conv3x3(mean,w5)+b5), w6)+b6)
// 32x32 output tile per block, fully fused in LDS (mean halo +/-2).
// Mean tile staged via async global->LDS.
// ===========================================================================
__global__ __launch_bounds__(256) void k2_att(
    const float* __restrict__ meanm, const float* __restrict__ w5,
    const float* __restrict__ b5, const float* __restrict__ w6,
    const float* __restrict__ b6, float* __restrict__ att) {
  __shared__ float sm[36 * 36];
  __shared__ float st[34 * 34 * 8];
  __shared__ float sw5[72], sb5[8], sw6[72];
  __shared__ float sb6;
  const int tid = threadIdx.x;
  const int b = blockIdx.z;
  const int y0 = blockIdx.y * 32, x0 = blockIdx.x * 32;
  const float* mp = meanm + (size_t)b * HW;

  if (tid < 72) sw5[tid] = w5[tid];
  else if (tid < 80) sb5[tid - 72] = b5[tid - 72];
  else if (tid < 152) sw6[tid - 80] = w6[tid - 80];
  else if (tid == 152) sb6 = b6[0];

  // zeros (ds) and async copies (ASYNCcnt) target disjoint LDS addresses
  for (int i = tid; i < 36 * 36; i += 256) {
    const int r = i / 36, c = i % 36;
    const int gy = y0 + r - 2, gx = x0 + c - 2;
    if (gy >= 0 && gy < 256 && gx >= 0 && gx < 256)
      ldgsts32(mp + gy * W_ + gx, &sm[i]);
    else
      sm[i] = 0.f;
  }
  async_wait_all();
  __syncthreads();
  for (int i = tid; i < 34 * 34 * 8; i += 256) {
    const int oc = i & 7;
    const int pos = i >> 3;
    const int pr = pos / 34, pc = pos % 34;
    float v = sb5[oc];
#pragma unroll
    for (int dy = 0; dy < 3; ++dy)
#pragma unroll
      for (int dx = 0; dx < 3; ++dx)
        v += sw5[oc * 9 + dy * 3 + dx] * sm[(pr + dy) * 36 + pc + dx];
    st[pos * 8 + oc] = fmaxf(v, 0.f);
  }
  __syncthreads();
  for (int i = tid; i < 32 * 32; i += 256) {
    const int pr = i / 32, pc = i % 32;
    float v = sb6;
#pragma unroll
    for (int ic = 0; ic < 8; ++ic)
#pragma unroll
      for (int dy = 0; dy < 3; ++dy)
#pragma unroll
        for (int dx = 0; dx < 3; ++dx)
          v += sw6[ic * 9 + dy * 3 + dx] * st[((pr + dy) * 34 + pc + dx) * 8 + ic];
    v = 1.0f / (1.0f + __expf(-v));
    att[(size_t)b * HW + (y0 + pr) * W_ + (x0 + pc)] = v;
  }
}

// ===========================================================================
// Kernel 3: out[b] = relu(conv3x3(G, w7f) + b7), G = F0[31-b] * a0[31-b],
// w7f[o][c] = w7[o][c] + w7[o][c+16] (channel duplication folded).
// F0 tile staged raw into LDS via async transposing gather ([pos][ch]);
// the a0 factor is applied at WMMA B-operand build time.
// Implicit GEMM: 9 taps x 4 K-steps x 2 M-halves of f32 WMMAs per 16-px group.
// ===========================================================================
__global__ __launch_bounds__(256) void k3_conv_out(
    const float* __restrict__ F0, const float* __restrict__ att,
    const float* __restrict__ w7, const float* __restrict__ b7,
    float* __restrict__ out) {
  __shared__ float sg[18 * 34 * 16];   // raw F0 tile [pos=row*34+col][ch]
  __shared__ float sw[9 * 32 * 16];    // folded weights [tap][o][c]
  __shared__ float sa[18 * 34];        // a0 tile
  __shared__ float sb[32];             // bias
  const int tid = threadIdx.x;
  const int bo = blockIdx.z;           // output batch
  const int bs = 31 - bo;              // source batch (jnp.flip)
  const int y0 = blockIdx.y * 16, x0 = blockIdx.x * 32;

  // fold w7 over duplicated channels
  for (int i = tid; i < 4608; i += 256) {
    const int tap = i >> 9;
    const int rem = i & 511;
    const int o = rem >> 4, c = rem & 15;
    sw[i] = w7[(o * 32 + c) * 9 + tap] + w7[(o * 32 + c + 16) * 9 + tap];
  }
  if (tid < 32) sb[tid] = b7[tid];

  // a0 tile: zero the out-of-image halo (ds), async-copy valid pixels
  const float* ap = att + (size_t)bs * HW;
  for (int p = tid; p < 18 * 34; p += 256) {
    const int r = p / 34, c = p % 34;
    const int gy = y0 + r - 1, gx = x0 + c - 1;
    if (gy >= 0 && gy < 256 && gx >= 0 && gx < 256)
      ldgsts32(ap + gy * W_ + gx, &sa[p]);
    else
      sa[p] = 0.f;
  }
  // F0 tile: zero out-of-image positions first (disjoint from async targets)
  const float* fp = F0 + (size_t)bs * 16 * HW;
  for (int p = tid; p < 18 * 34; p += 256) {
    const int r = p / 34, c = p % 34;
    const int gy = y0 + r - 1, gx = x0 + c - 1;
    if (!(gy >= 0 && gy < 256 && gx >= 0 && gx < 256)) {
#pragma unroll
      for (int ch = 0; ch < 16; ++ch) sg[p * 16 + ch] = 0.f;
    }
  }
  // item order (c, pos) so each async instr reads contiguous global memory
  for (int i = tid; i < 16 * 612; i += 256) {
    const int c = i / 612;             // channel plane
    const int p = i % 612;             // position (col fastest -> coalesced)
    const int r = p / 34, cc = p % 34;
    const int gy = y0 + r - 1, gx = x0 + cc - 1;
    if (gy >= 0 && gy < 256 && gx >= 0 && gx < 256)
      ldgsts32(fp + (size_t)c * HW + gy * W_ + gx, &sg[p * 16 + c]);
  }
  async_wait_all();
  __syncthreads();

  const int lane = tid & 31;
  const int wave = tid >> 5;
  const int half = lane >> 4;
  const int l16 = tid & 15;

  // accumulators: 4 pixel-groups x (out-ch 0-15, 16-31), bias preloaded
  v8f cinit[2];
#pragma unroll
  for (int h = 0; h < 2; ++h)
#pragma unroll
    for (int r = 0; r < 8; ++r) cinit[h][r] = sb[h * 16 + r + half * 8];
  v8f acc[4][2];
#pragma unroll
  for (int i = 0; i < 4; ++i) { acc[i][0] = cinit[0]; acc[i][1] = cinit[1]; }

  for (int tap = 0; tap < 9; ++tap) {
    const int ty = tap / 3, tx = tap % 3;
    // a0 factor per (tap, group): same pixel as the B-operand column
    float av[4];
#pragma unroll
    for (int i = 0; i < 4; ++i) {
      const int g = wave + i * 8;
      const int r = g >> 1, chf = g & 1;
      av[i] = sa[(r + ty) * 34 + chf * 16 + l16 + tx];
    }
#pragma unroll
    for (int kb = 0; kb < 4; ++kb) {
      const int kk = 4 * kb + half * 2;    // K rows for this lane half
      v2f wa0, wa1;
      wa0.x = sw[tap * 512 + (l16) * 16 + kk];
      wa0.y = sw[tap * 512 + (l16) * 16 + kk + 1];
      wa1.x = sw[tap * 512 + (16 + l16) * 16 + kk];
      wa1.y = sw[tap * 512 + (16 + l16) * 16 + kk + 1];
#pragma unroll
      for (int i = 0; i < 4; ++i) {
        const int g = wave + i * 8;        // group: row r, col-half chf
        const int r = g >> 1, chf = g & 1;
        const int gcol = chf * 16 + l16 + tx;
        const float* bp = &sg[((r + ty) * 34 + gcol) * 16 + kk];
        v2f bv;
        bv.x = bp[0] * av[i];
        bv.y = bp[1] * av[i];
        acc[i][0] = wmma4(wa0, bv, acc[i][0]);
        acc[i][1] = wmma4(wa1, bv, acc[i][1]);
      }
    }
  }

  // store: reg rr of half h -> out channel h*16 + rr + half*8, 16 px coalesced
  float* op = out + (size_t)bo * 32 * HW;
#pragma unroll
  for (int i = 0; i < 4; ++i) {
    const int g = wave + i * 8;
    const int r = g >> 1, chf = g & 1;
    const int orow = y0 + r, ocol = x0 + chf * 16 + l16;
#pragma unroll
    for (int h = 0; h < 2; ++h)
#pragma unroll
      for (int rr = 0; rr < 8; ++rr) {
        const int o = h * 16 + rr + half * 8;
        op[(size_t)o * HW + orow * W_ + ocol] = fmaxf(acc[i][h][rr], 0.f);
      }
  }
}

// ===========================================================================
extern "C" void kernel_launch(void* const* d_in, const int* in_sizes, int n_in,
                              void* d_out, int out_size, void* d_ws,
                              size_t ws_size, hipStream_t stream) {
  const float* x  = (const float*)d_in[0];
  const float* w1 = (const float*)d_in[1];
  const float* b1 = (const float*)d_in[2];
  // d_in[3..8] = w2,b2,w3,b3,w4,b4 : dead (argsort over size-1 axis picks F0)
  const float* w5 = (const float*)d_in[9];
  const float* b5 = (const float*)d_in[10];
  const float* w6 = (const float*)d_in[11];
  const float* b6 = (const float*)d_in[12];
  const float* w7 = (const float*)d_in[13];
  const float* b7 = (const float*)d_in[14];
  float* outp = (float*)d_out;

  float* F0    = (float*)d_ws;                 // 32*16*65536 floats (134 MB)
  float* meanm = F0 + (size_t)B_ * 16 * HW;    // 32*65536 floats (8.4 MB)
  float* attm  = meanm + (size_t)B_ * HW;      // 32*65536 floats (8.4 MB)

  // K1: 32 images * 256 rows / 8 rows-per-block
  k1_f0_mean<<<1024, 256, 0, stream>>>(x, w1, b1, F0, meanm);
  // K2: 32 images * 8x8 tiles of 32x32
  k2_att<<<dim3(8, 8, 32), 256, 0, stream>>>(meanm, w5, b5, w6, b6, attm);
  // K3: 32 images * 8 col-tiles(32) * 16 row-tiles(16)
  k3_conv_out<<<dim3(8, 16, 32), 256, 0, stream>>>(F0, attm, w7, b7, outp);
}